// Preprocessing_59682865545708
// MI455X (gfx1250) — compile-verified
//
#include <hip/hip_runtime.h>

// B=65536 frames, N=256 atoms, A=64 align atoms, M=128 output atoms.
// out[b, m*3+j] = ((traj[b,nn[m],:] - centroid_b) @ R_b)[j],  R_b = Kabsch(traj[b,align], ref[align])
//
// Bandwidth-bound (~300MB @ 23.3TB/s ~= 13us). One wave32 per frame (8 frames/wave serial).
// Covariance + centroid via chained V_WMMA_F32_16X16X4_F32 (zero cross-lane reductions),
// rotation via QCP quaternion (Newton on quartic + adjugate eigenvector).
// Frame staging via GLOBAL_LOAD_ASYNC_TO_LDS_B128 with double buffering when available.

typedef __attribute__((ext_vector_type(2))) float v2f;
typedef __attribute__((ext_vector_type(8))) float v8f;

#define WAVES_PER_WG 8
#define FPW 8         // frames per wave
#define SLAB_F 776    // 768 frame floats + zero pad word (+ pad to 16B multiple)

#if defined(__has_builtin)
#if __has_builtin(__builtin_amdgcn_global_load_async_to_lds_b128) && \
    __has_builtin(__builtin_amdgcn_s_wait_asynccnt)
#define USE_ASYNC_LDS 1
#endif
#endif

__device__ __forceinline__ float rl(float v, int lane) {
  return __builtin_bit_cast(float, __builtin_amdgcn_readlane(__builtin_bit_cast(int, v), lane));
}

__device__ __forceinline__ float det3(float a, float b, float c,
                                      float d, float e, float f,
                                      float g, float h, float i) {
  return a * (e * i - f * h) - b * (d * i - f * g) + c * (d * h - e * g);
}

#ifdef USE_ASYNC_LDS
// builtin expects pointer-to-int4 (vector_size(16)); give it correctly AS-qualified ones
// (named-AS -> generic conversions are implicit, so this matches either prototype form).
typedef int v4i __attribute__((vector_size(16)));
typedef __attribute__((address_space(1))) v4i gv4i;
typedef __attribute__((address_space(3))) v4i lv4i;

// issue 6 async b128 copies: one 3KB frame -> LDS slab (imm offset applies to both sides)
__device__ __forceinline__ void stage_async(const float* gsrc, float* lbuf, int lane) {
  gv4i* g = (gv4i*)(gsrc) + lane;
  lv4i* l = (lv4i*)(lbuf) + lane;
  __builtin_amdgcn_global_load_async_to_lds_b128(g, l, 0, 0);
  __builtin_amdgcn_global_load_async_to_lds_b128(g, l, 512, 0);
  __builtin_amdgcn_global_load_async_to_lds_b128(g, l, 1024, 0);
  __builtin_amdgcn_global_load_async_to_lds_b128(g, l, 1536, 0);
  __builtin_amdgcn_global_load_async_to_lds_b128(g, l, 2048, 0);
  __builtin_amdgcn_global_load_async_to_lds_b128(g, l, 2560, 0);
}
#endif

__global__ __launch_bounds__(256) void kabsch_align_kernel(
    const float* __restrict__ traj, const float* __restrict__ ref_pos,
    const int* __restrict__ align_idx, const int* __restrict__ nn_idx,
    float* __restrict__ out) {
  __shared__ __align__(16) float ldsbuf[WAVES_PER_WG * 2 * SLAB_F];
  const int lane = threadIdx.x & 31;
  const int wv = threadIdx.x >> 5;
  float* buf0 = ldsbuf + wv * 2 * SLAB_F;
  float* buf1 = buf0 + SLAB_F;

  const int i = lane & 15;             // A-frag row (coordinate) / B-frag column
  const int k0 = (lane < 16) ? 0 : 2;  // K pair owned by this half-wave

  // zero pad word used by A-fragment lanes with i>=3 (all lanes store same value: benign)
  buf0[768] = 0.0f;
  buf1[768] = 0.0f;

  // ---------------- one-time per-wave setup ----------------
  float rmx = 0.f, rmy = 0.f, rmz = 0.f;
  for (int n = 0; n < 64; ++n) {
    int a = align_idx[n];
    rmx += ref_pos[a * 3 + 0];
    rmy += ref_pos[a * 3 + 1];
    rmz += ref_pos[a * 3 + 2];
  }
  rmx *= 0.015625f; rmy *= 0.015625f; rmz *= 0.015625f;
  const float rmi = (i == 0) ? rmx : ((i == 1) ? rmy : rmz);

  // B fragments: B[k][j] = refc[atom 4t+k][j] (j<3), B[k][3] = 1 (centroid column)
  // f32 4x16 B layout: VGPR0 = row k0 (lanes0-15:K0, lanes16-31:K2), VGPR1 = row k0+1
  v2f bfrag[16];
  int aoff0[16], aoff1[16];
#pragma unroll
  for (int t = 0; t < 16; ++t) {
    int a0 = align_idx[4 * t + k0];
    int a1 = align_idx[4 * t + k0 + 1];
    aoff0[t] = (i < 3) ? (a0 * 3 + i) : 768;  // lanes i>=3 read the zero pad word
    aoff1[t] = (i < 3) ? (a1 * 3 + i) : 768;
    float b0, b1;
    if (i < 3) {
      b0 = ref_pos[a0 * 3 + i] - rmi;
      b1 = ref_pos[a1 * 3 + i] - rmi;
    } else if (i == 3) {
      b0 = 1.0f; b1 = 1.0f;
    } else {
      b0 = 0.0f; b1 = 0.0f;
    }
    bfrag[t].x = b0;
    bfrag[t].y = b1;
  }

  // each lane owns 4 output atoms: nn[4*lane .. 4*lane+3]
  int nnoff[4];
#pragma unroll
  for (int k = 0; k < 4; ++k) nnoff[k] = nn_idx[4 * lane + k] * 3;

  const long frame0 = ((long)blockIdx.x * WAVES_PER_WG + wv) * FPW;

#ifdef USE_ASYNC_LDS
  stage_async(traj + frame0 * 768, buf0, lane);  // prefetch frame 0
#endif

  for (int f = 0; f < FPW; ++f) {
    const long frame = frame0 + f;

#ifdef USE_ASYNC_LDS
    float* slab = (f & 1) ? buf1 : buf0;
    if (f + 1 < FPW) {
      stage_async(traj + (frame + 1) * 768, ((f + 1) & 1) ? buf1 : buf0, lane);
      __builtin_amdgcn_s_wait_asynccnt(6);  // in-order: <=6 outstanding => frame f resident
    } else {
      __builtin_amdgcn_s_wait_asynccnt(0);
    }
#else
    float* slab = buf0;
    const float4* fsrc = (const float4*)(traj + frame * 768);
    float4* slab4 = (float4*)slab;
#pragma unroll
    for (int k = 0; k < 6; ++k) slab4[lane + 32 * k] = fsrc[lane + 32 * k];
#endif

    // ---- gather all A-fragment values first (unconditional ds_load, pipelined) ----
    float av0[16], av1[16];
#pragma unroll
    for (int t = 0; t < 16; ++t) {
      av0[t] = slab[aoff0[t]];
      av1[t] = slab[aoff1[t]];
    }

    // ---- covariance S[i][j] = sum_n x[n,i]*refc[n,j] + centroid sums via chained WMMA ----
    v8f acc = {0.f, 0.f, 0.f, 0.f, 0.f, 0.f, 0.f, 0.f};
#pragma unroll
    for (int t = 0; t < 16; ++t) {
      v2f a;
      a.x = av0[t];
      a.y = av1[t];
      acc = __builtin_amdgcn_wmma_f32_16x16x4_f32(false, a, false, bfrag[t],
                                                  (short)0, acc, false, false);
    }

    // ---- extract S (3x3) and centroid: D[i][j] lives in VGPR i, lane j ----
    float g00 = rl(acc[0], 0), g01 = rl(acc[0], 1), g02 = rl(acc[0], 2);
    float g10 = rl(acc[1], 0), g11 = rl(acc[1], 1), g12 = rl(acc[1], 2);
    float g20 = rl(acc[2], 0), g21 = rl(acc[2], 1), g22 = rl(acc[2], 2);
    float cx = rl(acc[0], 3) * 0.015625f;
    float cy = rl(acc[1], 3) * 0.015625f;
    float cz = rl(acc[2], 3) * 0.015625f;
    // ref is centered, so S with raw x equals S with centered x (sum refc = 0).

    // ---- QCP: Horn matrix H[u][v] = sum ref_u * x_v = g[v][u] ----
    float Hxx = g00, Hxy = g10, Hxz = g20;
    float Hyx = g01, Hyy = g11, Hyz = g21;
    float Hzx = g02, Hzy = g12, Hzz = g22;

    float N00 = Hxx + Hyy + Hzz;
    float N01 = Hyz - Hzy;
    float N02 = Hzx - Hxz;
    float N03 = Hxy - Hyx;
    float N11 = Hxx - Hyy - Hzz;
    float N12 = Hxy + Hyx;
    float N13 = Hzx + Hxz;
    float N22 = -Hxx + Hyy - Hzz;
    float N23 = Hyz + Hzy;
    float N33 = -Hxx - Hyy + Hzz;

    // char poly det(lam*I - N) = lam^4 + e2*lam^2 - e3*lam + e4   (tr N = 0)
    float e2 = N00 * N11 - N01 * N01 + N00 * N22 - N02 * N02 + N00 * N33 - N03 * N03 +
               N11 * N22 - N12 * N12 + N11 * N33 - N13 * N13 + N22 * N33 - N23 * N23;
    float e3 = det3(N00, N01, N02, N01, N11, N12, N02, N12, N22) +
               det3(N00, N01, N03, N01, N11, N13, N03, N13, N33) +
               det3(N00, N02, N03, N02, N22, N23, N03, N23, N33) +
               det3(N11, N12, N13, N12, N22, N23, N13, N23, N33);
    float e4 = N00 * det3(N11, N12, N13, N12, N22, N23, N13, N23, N33) -
               N01 * det3(N01, N12, N13, N02, N22, N23, N03, N23, N33) +
               N02 * det3(N01, N11, N13, N02, N12, N23, N03, N13, N33) -
               N03 * det3(N01, N11, N12, N02, N12, N22, N03, N13, N23);

    // Newton from an upper bound: lam_max <= sqrt(3 * tr(S S^T))
    float sumsq = Hxx * Hxx + Hxy * Hxy + Hxz * Hxz + Hyx * Hyx + Hyy * Hyy +
                  Hyz * Hyz + Hzx * Hzx + Hzy * Hzy + Hzz * Hzz;
    float lam = __builtin_sqrtf(3.0f * sumsq) + 1e-7f;
    for (int it = 0; it < 10; ++it) {
      float l2 = lam * lam;
      float P = (l2 + e2) * l2 - e3 * lam + e4;
      float dP = (4.0f * l2 + 2.0f * e2) * lam - e3;
      if (__builtin_fabsf(dP) < 1e-12f) break;
      lam -= P / dP;
    }

    // eigenvector of (N - lam I): adjugate row (cofactors), with fallbacks
    float b00 = N00 - lam, b11 = N11 - lam, b22 = N22 - lam, b33 = N33 - lam;
    float b01 = N01, b02 = N02, b03 = N03, b12 = N12, b13 = N13, b23 = N23;

    float bn = b00 * b00 + b11 * b11 + b22 * b22 + b33 * b33 +
               2.0f * (b01 * b01 + b02 * b02 + b03 * b03 + b12 * b12 + b13 * b13 + b23 * b23);
    float thr = 1e-6f * bn * bn * bn + 1e-30f;

    float q0 = det3(b11, b12, b13, b12, b22, b23, b13, b23, b33);
    float q1 = -det3(b01, b12, b13, b02, b22, b23, b03, b23, b33);
    float q2 = det3(b01, b11, b13, b02, b12, b23, b03, b13, b33);
    float q3 = -det3(b01, b11, b12, b02, b12, b22, b03, b13, b23);
    float qq = q0 * q0 + q1 * q1 + q2 * q2 + q3 * q3;
    if (qq < thr) {
      q0 = -det3(b01, b02, b03, b12, b22, b23, b13, b23, b33);
      q1 = det3(b00, b02, b03, b02, b22, b23, b03, b23, b33);
      q2 = -det3(b00, b01, b03, b02, b12, b23, b03, b13, b33);
      q3 = det3(b00, b01, b02, b02, b12, b22, b03, b13, b23);
      qq = q0 * q0 + q1 * q1 + q2 * q2 + q3 * q3;
      if (qq < thr) {
        q0 = det3(b01, b02, b03, b11, b12, b13, b13, b23, b33);
        q1 = -det3(b00, b02, b03, b01, b12, b13, b03, b23, b33);
        q2 = det3(b00, b01, b03, b01, b11, b13, b03, b13, b33);
        q3 = -det3(b00, b01, b02, b01, b11, b12, b03, b13, b23);
        qq = q0 * q0 + q1 * q1 + q2 * q2 + q3 * q3;
        if (qq < thr) { q0 = 1.0f; q1 = 0.0f; q2 = 0.0f; q3 = 0.0f; qq = 1.0f; }
      }
    }
    float qs = 1.0f / __builtin_sqrtf(qq);
    float w = q0 * qs, x = q1 * qs, y = q2 * qs, z = q3 * qs;

    // rot = R(q): aligned_j = sum_i (p_i - c_i) * R[i][j]
    float r00 = 1.0f - 2.0f * (y * y + z * z);
    float r01 = 2.0f * (x * y - w * z);
    float r02 = 2.0f * (x * z + w * y);
    float r10 = 2.0f * (x * y + w * z);
    float r11 = 1.0f - 2.0f * (x * x + z * z);
    float r12 = 2.0f * (y * z - w * x);
    float r20 = 2.0f * (x * z - w * y);
    float r21 = 2.0f * (y * z + w * x);
    float r22 = 1.0f - 2.0f * (x * x + y * y);

    // ---- apply rotation to this lane's 4 nn atoms; coalesced b128 stores ----
    float o[12];
#pragma unroll
    for (int k = 0; k < 4; ++k) {
      float px = slab[nnoff[k] + 0] - cx;
      float py = slab[nnoff[k] + 1] - cy;
      float pz = slab[nnoff[k] + 2] - cz;
      o[3 * k + 0] = px * r00 + py * r10 + pz * r20;
      o[3 * k + 1] = px * r01 + py * r11 + pz * r21;
      o[3 * k + 2] = px * r02 + py * r12 + pz * r22;
    }
    float4* ob = (float4*)(out + frame * 384 + lane * 12);
    ob[0] = make_float4(o[0], o[1], o[2], o[3]);
    ob[1] = make_float4(o[4], o[5], o[6], o[7]);
    ob[2] = make_float4(o[8], o[9], o[10], o[11]);
  }
}

extern "C" void kernel_launch(void* const* d_in, const int* in_sizes, int n_in,
                              void* d_out, int out_size, void* d_ws, size_t ws_size,
                              hipStream_t stream) {
  (void)n_in; (void)out_size; (void)d_ws; (void)ws_size;
  const float* traj = (const float*)d_in[0];
  const float* ref_pos = (const float*)d_in[1];
  const int* align_idx = (const int*)d_in[2];
  const int* nn_idx = (const int*)d_in[3];
  float* out = (float*)d_out;

  const int B = in_sizes[0] / (256 * 3);      // 65536
  const int grid = B / (WAVES_PER_WG * FPW);  // 1024 workgroups
  kabsch_align_kernel<<<grid, WAVES_PER_WG * 32, 0, stream>>>(traj, ref_pos, align_idx,
                                                              nn_idx, out);
}